// GATReduce_33114197852456
// MI455X (gfx1250) — compile-verified
//
#include <hip/hip_runtime.h>
#include <stdint.h>

#define D_FEAT 48
#define TILE 256
#define NTHREADS 256

// ---- CDNA5 async global->LDS copy (ASYNCcnt path), inline asm -------------
// VFLAT encoding: global_load_async_to_lds_b128 vdst(LDS byte addr), v[addr], off
__device__ __forceinline__ void async_copy_b128(void* lds_dst, const void* gsrc) {
  unsigned loff = (unsigned)(uintptr_t)lds_dst;          // LDS byte offset = addr[31:0]
  unsigned long long ga = (unsigned long long)(uintptr_t)gsrc;
  asm volatile("global_load_async_to_lds_b128 %0, %1, off"
               :: "v"(loff), "v"(ga) : "memory");
}

__device__ __forceinline__ void wait_async0() {
#if __has_builtin(__builtin_amdgcn_s_wait_asynccnt)
  __builtin_amdgcn_s_wait_asynccnt(0);
#else
  asm volatile("s_wait_asynccnt 0" ::: "memory");
#endif
}

// ---- float atomic max via ordered-int trick (init must be -inf) -----------
__device__ __forceinline__ void atomicMaxF(float* addr, float val) {
  if (val >= 0.0f) {
    atomicMax((int*)addr, __float_as_int(val));
  } else {
    atomicMin((unsigned int*)addr, (unsigned int)__float_as_int(val));
  }
}

// ---- kernel 0: init out / segmax / denom ----------------------------------
__global__ void k_init(float* __restrict__ out, float* __restrict__ segmax,
                       float* __restrict__ denom, int nd, int n) {
  int i = blockIdx.x * blockDim.x + threadIdx.x;
  if (i < nd) out[i] = 0.0f;
  if (i < n) {
    segmax[i] = -__builtin_inff();
    denom[i] = 0.0f;
  }
}

// ---- kernel 1: per-node max (wave32 segmented max-scan, sorted ids) -------
__global__ void k_segmax(const float* __restrict__ a, const int* __restrict__ sid,
                         float* __restrict__ segmax, int E) {
  int e = blockIdx.x * blockDim.x + threadIdx.x;
  if (e >= E) return;                       // E % 256 == 0 for this problem
  float v = a[e];
  int s = sid[e];
  int lane = threadIdx.x & 31;
#pragma unroll
  for (int off = 1; off < 32; off <<= 1) {
    float ov = __shfl_up(v, off, 32);
    int os = __shfl_up(s, off, 32);
    if (lane >= off && os == s) v = fmaxf(v, ov);
  }
  int ns = __shfl_down(s, 1, 32);
  if (lane == 31 || ns != s || e == E - 1) atomicMaxF(&segmax[s], v);
}

// ---- kernel 2: per-node sum of exp(a - max) (segmented sum-scan) ----------
__global__ void k_denom(const float* __restrict__ a, const int* __restrict__ sid,
                        const float* __restrict__ segmax, float* __restrict__ denom,
                        int E) {
  int e = blockIdx.x * blockDim.x + threadIdx.x;
  if (e >= E) return;
  int s = sid[e];
  float v = __expf(a[e] - segmax[s]);
  int lane = threadIdx.x & 31;
#pragma unroll
  for (int off = 1; off < 32; off <<= 1) {
    float ov = __shfl_up(v, off, 32);
    int os = __shfl_up(s, off, 32);
    if (lane >= off && os == s) v += ov;
  }
  int ns = __shfl_down(s, 1, 32);
  if (lane == 31 || ns != s || e == E - 1) atomicAdd(&denom[s], v);
}

// ---- kernel 3: main pass — async-stage ft tile, weighted scatter-reduce ---
__global__ void __launch_bounds__(NTHREADS)
k_accum(const float* __restrict__ a, const float* __restrict__ ft,
        const int* __restrict__ sid, const float* __restrict__ segmax,
        const float* __restrict__ denom, float* __restrict__ out, int E) {
  __shared__ float ft_s[TILE * D_FEAT];   // 48 KB tile
  __shared__ float w_s[TILE];
  __shared__ int s_s[TILE];

  const int t = threadIdx.x;
  const long base = (long)blockIdx.x * TILE;

  // Stage A: async global->LDS staging of the 256x48 f32 tile.
  // 3072 float4 transfers; 12 fully-coalesced b128 async copies per thread.
  {
    const long tile_f4 = base * (D_FEAT / 4);
    const long total_f4 = (long)E * (D_FEAT / 4);
#pragma unroll
    for (int k = 0; k < (TILE * D_FEAT / 4) / NTHREADS; ++k) {  // 12
      int idx = t + k * NTHREADS;         // tile-local float4 index
      long gf4 = tile_f4 + idx;
      if (gf4 >= total_f4) gf4 = tile_f4; // tail clamp (unused when E%TILE==0)
      async_copy_b128(&ft_s[idx * 4], &ft[gf4 * 4]);
    }
  }

  // Stage B (overlaps async copies): per-edge softmax weight.
  {
    long e = base + t;
    float w = 0.0f;
    int s = 0;
    if (e < E) {
      s = sid[e];
      w = __expf(a[e] - segmax[s]) / denom[s];
    }
    w_s[t] = w;
    s_s[t] = s;   // padded lanes: w==0 -> contributes nothing
  }

  wait_async0();       // our wave's async copies landed in LDS
  __syncthreads();     // all waves' copies + weights visible

  // Stage C: 16 edge sub-chunks x 16 feature groups (3 features / thread).
  // Sorted ids -> accumulate runs in registers, atomic flush at boundaries.
  const int sub = t >> 4;   // edges [sub*16, sub*16+16)
  const int g = t & 15;     // features [3g, 3g+3)
  const int fo = g * 3;
  const int ebeg = sub * 16;

  float a0 = 0.f, a1 = 0.f, a2 = 0.f;
  int cur = s_s[ebeg];
#pragma unroll
  for (int i = 0; i < 16; ++i) {
    int el = ebeg + i;
    int s = s_s[el];
    if (s != cur) {
      float* o = &out[(long)cur * D_FEAT + fo];
      atomicAdd(o + 0, a0);
      atomicAdd(o + 1, a1);
      atomicAdd(o + 2, a2);
      a0 = a1 = a2 = 0.f;
      cur = s;
    }
    float w = w_s[el];
    const float* r = &ft_s[el * D_FEAT + fo];
    a0 = fmaf(w, r[0], a0);
    a1 = fmaf(w, r[1], a1);
    a2 = fmaf(w, r[2], a2);
  }
  {
    float* o = &out[(long)cur * D_FEAT + fo];
    atomicAdd(o + 0, a0);
    atomicAdd(o + 1, a1);
    atomicAdd(o + 2, a2);
  }
}

// ---- launcher --------------------------------------------------------------
extern "C" void kernel_launch(void* const* d_in, const int* in_sizes, int n_in,
                              void* d_out, int out_size, void* d_ws, size_t ws_size,
                              hipStream_t stream) {
  const float* a = (const float*)d_in[0];
  const float* ft = (const float*)d_in[1];
  const int* sid = (const int*)d_in[2];
  const int E = in_sizes[0];
  const int N = out_size / D_FEAT;

  float* out = (float*)d_out;
  float* segmax = (float*)d_ws;      // N floats
  float* denom = segmax + N;         // N floats  (ws needs 2*N*4 = 400 KB)

  const int nd = N * D_FEAT;
  k_init<<<(nd + 255) / 256, 256, 0, stream>>>(out, segmax, denom, nd, N);

  const int eb = (E + 255) / 256;
  k_segmax<<<eb, 256, 0, stream>>>(a, sid, segmax, E);
  k_denom<<<eb, 256, 0, stream>>>(a, sid, segmax, denom, E);

  const int tb = (E + TILE - 1) / TILE;
  k_accum<<<tb, NTHREADS, 0, stream>>>(a, ft, sid, segmax, denom, out, E);
}